// GMGSRegressionLoss4_32289564131834
// MI455X (gfx1250) — compile-verified
//
#include <hip/hip_runtime.h>
#include <math.h>

typedef float v2f __attribute__((ext_vector_type(2)));
typedef float v8f __attribute__((ext_vector_type(8)));

#define T_DIM    24
#define NBLOCKS  1024
#define NTHREADS 256
#define NWAVES   (NTHREADS / 32)

__device__ __forceinline__ int to_class(float v) {
    return (int)(v >= 0.0f) + (int)(v >= 1.0f) + (int)(v >= 2.0f);
}

// Exact f32 reduction of one value across the 32-lane wave using
// V_WMMA_F32_16X16X4_F32 with an all-ones B matrix:
//   A[m][k]: lane L supplies A[L%16][L<16 ? 0 : 2] (second A VGPR = 0)
//   D[m][n] = sum_k A[m][k] = v_m + v_{m+16}  (broadcast over n)
//   lanes 0-15 hold row-sums s_0..s_7 in d[0..7], lanes 16-31 hold s_8..s_15,
//   so total = (d0+..+d7) + shfl_xor(.,16). All lanes return the full sum.
__device__ __forceinline__ float wave_reduce_wmma(float v) {
    v2f a = {v, 0.0f};
    v2f b = {1.0f, 1.0f};
    v8f c = {};
    v8f d = __builtin_amdgcn_wmma_f32_16x16x4_f32(
        /*neg_a=*/false, a, /*neg_b=*/false, b,
        /*c_mod=*/(short)0, c, /*reuse_a=*/false, /*reuse_b=*/false);
    float t = ((d[0] + d[1]) + (d[2] + d[3])) + ((d[4] + d[5]) + (d[6] + d[7]));
    t += __shfl_xor(t, 16, 32);
    return t;
}

__global__ void __launch_bounds__(NTHREADS)
gmgs_partial_kernel(const float* __restrict__ pred,
                    const float* __restrict__ tru,
                    const float* __restrict__ smat,
                    float2* __restrict__ partials,
                    int rows) {
    __shared__ float smw[16];
    __shared__ float sqW[NWAVES];
    __shared__ float wW[NWAVES];

    // Normalized exp(-score_matrix) into LDS (16 values; cost negligible).
    if (threadIdx.x < 16) {
        float s = 0.0f;
        #pragma unroll
        for (int j = 0; j < 16; ++j) s += expf(-smat[j]);
        smw[threadIdx.x] = expf(-smat[threadIdx.x]) / s;
    }
    __syncthreads();

    float sq = 0.0f;   // partial sum of (pred-true)^2
    float ws = 0.0f;   // partial sum of gmgs weights

    const int stride = gridDim.x * blockDim.x;
    for (int r = blockIdx.x * blockDim.x + threadIdx.x; r < rows; r += stride) {
        const float4* p4 = (const float4*)(pred + (size_t)r * T_DIM);
        const float4* q4 = (const float4*)(tru  + (size_t)r * T_DIM);
        float mp = -INFINITY, mt = -INFINITY;
        float s = 0.0f;
        #pragma unroll
        for (int k = 0; k < T_DIM / 4; ++k) {   // 6 x float4 = one 96B row
            float4 a = p4[k];
            float4 b = q4[k];
            mp = fmaxf(mp, fmaxf(fmaxf(a.x, a.y), fmaxf(a.z, a.w)));
            mt = fmaxf(mt, fmaxf(fmaxf(b.x, b.y), fmaxf(b.z, b.w)));
            float dx = a.x - b.x, dy = a.y - b.y;
            float dz = a.z - b.z, dw = a.w - b.w;
            s += dx * dx + dy * dy + dz * dz + dw * dw;
        }
        sq += s;
        ws += smw[to_class(mt) * 4 + to_class(mp)];
    }

    // Wave-level reduction via WMMA (EXEC all-1s here: full block, post-loop).
    float sqTot = wave_reduce_wmma(sq);
    float wTot  = wave_reduce_wmma(ws);

    const int wid = threadIdx.x >> 5;
    if ((threadIdx.x & 31) == 0) { sqW[wid] = sqTot; wW[wid] = wTot; }
    __syncthreads();
    if (threadIdx.x == 0) {
        float a = 0.0f, b = 0.0f;
        #pragma unroll
        for (int i = 0; i < NWAVES; ++i) { a += sqW[i]; b += wW[i]; }
        partials[blockIdx.x] = make_float2(a, b);
    }
}

__global__ void __launch_bounds__(NTHREADS)
gmgs_final_kernel(const float2* __restrict__ partials,
                  float* __restrict__ out,
                  int nparts, int rows) {
    __shared__ float sqW[NWAVES];
    __shared__ float wW[NWAVES];

    float sq = 0.0f, ws = 0.0f;
    for (int i = threadIdx.x; i < nparts; i += blockDim.x) {
        float2 p = partials[i];
        sq += p.x;
        ws += p.y;
    }
    float sqTot = wave_reduce_wmma(sq);
    float wTot  = wave_reduce_wmma(ws);

    const int wid = threadIdx.x >> 5;
    if ((threadIdx.x & 31) == 0) { sqW[wid] = sqTot; wW[wid] = wTot; }
    __syncthreads();
    if (threadIdx.x == 0) {
        float a = 0.0f, b = 0.0f;
        #pragma unroll
        for (int i = 0; i < NWAVES; ++i) { a += sqW[i]; b += wW[i]; }
        out[0] = a * b / (float)rows;
    }
}

extern "C" void kernel_launch(void* const* d_in, const int* in_sizes, int n_in,
                              void* d_out, int out_size, void* d_ws, size_t ws_size,
                              hipStream_t stream) {
    const float* pred = (const float*)d_in[0];   // (B, 24, 1) f32
    const float* tru  = (const float*)d_in[1];   // (B, 24, 1) f32
    const float* smat = (const float*)d_in[2];   // (4, 4) f32
    const int rows = in_sizes[0] / T_DIM;

    float2* partials = (float2*)d_ws;            // NBLOCKS * 8 bytes = 8 KB

    gmgs_partial_kernel<<<NBLOCKS, NTHREADS, 0, stream>>>(pred, tru, smat, partials, rows);
    gmgs_final_kernel<<<1, NTHREADS, 0, stream>>>(partials, (float*)d_out, NBLOCKS, rows);
}